// NNUE_43774306680937
// MI455X (gfx1250) — compile-verified
//
#include <hip/hip_runtime.h>
#include <math.h>

typedef __attribute__((ext_vector_type(2))) float v2f;
typedef __attribute__((ext_vector_type(8))) float v8f;

#define WAVES_PER_BLOCK 4
#define ROWS_PER_WAVE   16
#define ROWS_PER_BLOCK  (WAVES_PER_BLOCK * ROWS_PER_WAVE)   // 64
#define NNZ     32
#define FT      64
#define XCOLS   96           // 64 folded-crelu + 32 dense
#define XSTRIDE 97           // odd stride -> conflict-free row access
#define H1      64
#define H2      32

__global__ __launch_bounds__(128) void nnue_fused_kernel(
    const int*   __restrict__ sparse_idx,  // [B, 32]
    const float* __restrict__ dense,       // [B, 32]
    const float* __restrict__ ft_w,        // [32768, 64]
    const float* __restrict__ ft_b,        // [64]
    const float* __restrict__ W1,          // [160, 64]
    const float* __restrict__ b1,          // [64]
    const float* __restrict__ W2,          // [64, 32]
    const float* __restrict__ b2,          // [32]
    const float* __restrict__ W3,          // [32]
    const float* __restrict__ b3,          // [1]
    float*       __restrict__ out)         // [B]
{
    __shared__ float s_x[WAVES_PER_BLOCK][ROWS_PER_WAVE][XSTRIDE];

    const int lane    = threadIdx.x & 31;
    const int wave    = threadIdx.x >> 5;
    const int rowbase = blockIdx.x * ROWS_PER_BLOCK + wave * ROWS_PER_WAVE;
    // Wave-uniform copy so the compiler can prove index addresses uniform
    // and fetch them through the scalar (SMEM) path instead of 32-lane VMEM.
    const int rowbase_s = __builtin_amdgcn_readfirstlane(rowbase);

    float (*xw)[XSTRIDE] = s_x[wave];

    // ---- dense features -> x[:, 64..95] ----
    #pragma unroll
    for (int i = 0; i < ROWS_PER_WAVE; ++i)
        xw[i][FT + lane] = dense[(rowbase + i) * 32 + lane];

    // ---- EmbeddingBag gather-sum + CReLU -> x[:, 0..63] ----
    // Indices are wave-uniform -> s_load; each lane gathers 2 of 64 columns
    // (float2) from the L2-resident 8MB table. This loop is the bottleneck.
    const int c0 = 2 * lane;
    const float fb0 = ft_b[c0], fb1 = ft_b[c0 + 1];
    #pragma unroll 1
    for (int r = 0; r < ROWS_PER_WAVE; ++r) {
        const int* srow = sparse_idx + (rowbase_s + r) * NNZ;  // uniform
        float ax = fb0, ay = fb1;
        #pragma unroll
        for (int j = 0; j < NNZ; ++j) {
            const int ix = srow[j];                             // SMEM scalar load
            const float2 v = *(const float2*)(ft_w + ix * FT + c0);
            ax += v.x;  ay += v.y;
        }
        xw[r][c0]     = fmaxf(ax, 0.0f);
        xw[r][c0 + 1] = fmaxf(ay, 0.0f);
    }
    __builtin_amdgcn_wave_barrier();   // order LDS writes vs. A-fragment reads

    const int mrow  = lane & 15;        // M (A/D row) or N (B/C col) index
    const int khalf = (lane >> 4) * 2;  // lanes 16-31 hold K+2,K+3 of each chunk

    // ========== GEMM1: [16,96] x folded-W1[96,64] + b1, ReLU ==========
    // x = [crelu, crelu, dense]  =>  x@W1 = crelu@(W1a+W1b) + dense@W1c
    v8f c[4];
    #pragma unroll
    for (int nt = 0; nt < 4; ++nt) {
        const float bv = b1[nt * 16 + mrow];
        c[nt] = (v8f){bv, bv, bv, bv, bv, bv, bv, bv};
    }
    #pragma unroll 4
    for (int kc = 0; kc < FT / 4; ++kc) {          // folded crelu chunks (K 0..63)
        const int kb = kc * 4 + khalf;
        v2f a;  a.x = xw[mrow][kb];  a.y = xw[mrow][kb + 1];
        #pragma unroll
        for (int nt = 0; nt < 4; ++nt) {
            const int n = nt * 16 + mrow;
            v2f b;
            b.x = W1[kb * H1 + n]       + W1[(kb + FT) * H1 + n];
            b.y = W1[(kb + 1) * H1 + n] + W1[(kb + 1 + FT) * H1 + n];
            c[nt] = __builtin_amdgcn_wmma_f32_16x16x4_f32(
                        false, a, false, b, (short)0, c[nt], false, false);
        }
    }
    #pragma unroll 4
    for (int kc = FT / 4; kc < XCOLS / 4; ++kc) {  // dense chunks (orig K 128..159)
        const int kb = kc * 4 + khalf;
        v2f a;  a.x = xw[mrow][kb];  a.y = xw[mrow][kb + 1];
        #pragma unroll
        for (int nt = 0; nt < 4; ++nt) {
            const int n = nt * 16 + mrow;
            v2f b;
            b.x = W1[(kb + FT) * H1 + n];
            b.y = W1[(kb + 1 + FT) * H1 + n];
            c[nt] = __builtin_amdgcn_wmma_f32_16x16x4_f32(
                        false, a, false, b, (short)0, c[nt], false, false);
        }
    }
    __builtin_amdgcn_wave_barrier();
    // ReLU + transpose (lane=N layout -> row-major LDS) into x[:, 0..63]
    #pragma unroll
    for (int nt = 0; nt < 4; ++nt) {
        #pragma unroll
        for (int j = 0; j < 8; ++j) {
            const int m = j + (lane >> 4) * 8;
            xw[m][nt * 16 + mrow] = fmaxf(c[nt][j], 0.0f);
        }
    }
    __builtin_amdgcn_wave_barrier();

    // ========== GEMM2: [16,64] x [64,32] + b2, ReLU ==========
    v8f d2[2];
    #pragma unroll
    for (int nt = 0; nt < 2; ++nt) {
        const float bv = b2[nt * 16 + mrow];
        d2[nt] = (v8f){bv, bv, bv, bv, bv, bv, bv, bv};
    }
    #pragma unroll
    for (int kc = 0; kc < H1 / 4; ++kc) {
        const int kb = kc * 4 + khalf;
        v2f a;  a.x = xw[mrow][kb];  a.y = xw[mrow][kb + 1];
        #pragma unroll
        for (int nt = 0; nt < 2; ++nt) {
            const int n = nt * 16 + mrow;
            v2f b;  b.x = W2[kb * H2 + n];  b.y = W2[(kb + 1) * H2 + n];
            d2[nt] = __builtin_amdgcn_wmma_f32_16x16x4_f32(
                         false, a, false, b, (short)0, d2[nt], false, false);
        }
    }
    __builtin_amdgcn_wave_barrier();
    // ReLU + store h2[16,32] into x[:, 0..31]
    #pragma unroll
    for (int nt = 0; nt < 2; ++nt) {
        #pragma unroll
        for (int j = 0; j < 8; ++j) {
            const int m = j + (lane >> 4) * 8;
            xw[m][nt * 16 + mrow] = fmaxf(d2[nt][j], 0.0f);
        }
    }
    __builtin_amdgcn_wave_barrier();

    // ========== Layer 3: [16,32] x [32,1] + b3, tanh ==========
    // Lane pair (l, l+16) splits the 32-wide dot; combine with shfl_xor.
    const int r3  = lane & 15;
    const int kb3 = (lane >> 4) * 16;
    float sum = 0.0f;
    #pragma unroll
    for (int k = 0; k < 16; ++k)
        sum += xw[r3][kb3 + k] * W3[kb3 + k];
    sum += __shfl_xor(sum, 16, 32);
    if (lane < 16)
        out[rowbase + r3] = tanhf(sum + b3[0]);
}

extern "C" void kernel_launch(void* const* d_in, const int* in_sizes, int n_in,
                              void* d_out, int out_size, void* d_ws, size_t ws_size,
                              hipStream_t stream) {
    const int*   sparse_idx = (const int*)  d_in[0];
    const float* dense      = (const float*)d_in[1];
    const float* ft_w       = (const float*)d_in[2];
    const float* ft_b       = (const float*)d_in[3];
    const float* W1         = (const float*)d_in[4];
    const float* b1         = (const float*)d_in[5];
    const float* W2         = (const float*)d_in[6];
    const float* b2         = (const float*)d_in[7];
    const float* W3         = (const float*)d_in[8];
    const float* b3         = (const float*)d_in[9];
    float* out = (float*)d_out;

    const int B = in_sizes[1] / 32;                 // dense_batch is [B, 32]
    dim3 grid(B / ROWS_PER_BLOCK), block(128);      // 4 wave32s, 16 rows each
    nnue_fused_kernel<<<grid, block, 0, stream>>>(
        sparse_idx, dense, ft_w, ft_b, W1, b1, W2, b2, W3, b3, out);
}